// VectorQuantizerMS_38319698215616
// MI455X (gfx1250) — compile-verified
//
#include <hip/hip_runtime.h>

typedef float v2f __attribute__((ext_vector_type(2)));
typedef float v8f __attribute__((ext_vector_type(8)));
typedef unsigned int v4u __attribute__((ext_vector_type(4)));
typedef int v4i __attribute__((ext_vector_type(4)));
typedef int v8i __attribute__((ext_vector_type(8)));

#define K_CODES   512
#define D_DIM     64
#define TILE_PX   64
#define CH_STRIDE 68            // dword stride per pixel row in LDS (bank padding, keeps 8B align)
#define N_TILES   4096          // 262144 pixels / 64
#define HWPLANE   4096          // H*W
#define CT_FLOATS 1024          // one code tile: 16 codes x 64 ch

// ---------------------------------------------------------------------------
// Kernel 1: codebook squared norms -> ws[0..511]; also warms embedding in L2.
// ---------------------------------------------------------------------------
__global__ __launch_bounds__(256) void vq_prep(const float* __restrict__ emb,
                                               float* __restrict__ enorm) {
    int k = blockIdx.x * blockDim.x + threadIdx.x;
    if (k < K_CODES) {
        const float* e = emb + k * D_DIM;
        float s = 0.f;
#pragma unroll
        for (int d = 0; d < D_DIM; ++d) { float v = e[d]; s += v * v; }
        enorm[k] = s;
    }
}

// ---------------------------------------------------------------------------
// TDM helper: issue tensor_load_to_lds of 4KB (1024 f32, contiguous) from
// `gaddr` to LDS byte offset `lds_off`. D# per CDNA5 ISA ch.8:
//   group0: count=1, lds_addr, global_addr[56:0], type=2
//   group1: data_size=2 (4B), tensor_dim0=1024, tensor_dim1=1,
//           tile_dim0=1024, tile_dim1=1, stride0=1024
// Issued once per calling wave; tracked by TENSORcnt; EXEC ignored.
// ---------------------------------------------------------------------------
__device__ __forceinline__ void tdm_load_tile(const float* gsrc, unsigned lds_off) {
    unsigned long long ga = (unsigned long long)(const void*)gsrc;
    v4u g0;
    g0.x = 1u;                                   // count=1, user descriptor
    g0.y = lds_off;                              // lds_addr (bytes)
    g0.z = (unsigned)(ga & 0xFFFFFFFFu);         // global_addr[31:0]
    g0.w = (unsigned)(ga >> 32) | (2u << 30);    // global_addr[56:32] | type=2

    v8i g1;
    g1[0] = (2 << 16);                           // data_size = 4 bytes
    g1[1] = (int)((CT_FLOATS & 0xFFFF) << 16);   // tensor_dim0[15:0]
    g1[2] = (1 << 16);                           // tensor_dim0[31:16]=0 | tensor_dim1[15:0]=1
    g1[3] = (int)((CT_FLOATS & 0xFFFF) << 16);   // tensor_dim1[31:16]=0 | tile_dim0=1024
    g1[4] = 1;                                   // tile_dim1=1 | tile_dim2=0
    g1[5] = CT_FLOATS;                           // tensor_dim0_stride[31:0]
    g1[6] = 0;                                   // stride0[47:32] | stride1[15:0]
    g1[7] = 0;                                   // stride1[47:16]

    v4i gz = {0, 0, 0, 0};
#if defined(__clang_major__) && (__clang_major__ >= 23)
    v8i gz8 = {0, 0, 0, 0, 0, 0, 0, 0};
    __builtin_amdgcn_tensor_load_to_lds(g0, g1, gz, gz, gz8, 0);
#else
    __builtin_amdgcn_tensor_load_to_lds(g0, g1, gz, gz, 0);
#endif
}

// ---------------------------------------------------------------------------
// Kernel 2: main VQ. One block = 64 pixels (one (b,h) row), 4 waves x 16 px.
// Codebook tiles double-buffered into LDS by the Tensor Data Mover (wave 0),
// overlapped with V_WMMA_F32_16X16X4_F32 compute on the previous tile.
// dist = ||e||^2 - 2*(X.E^T); the row-constant ||x||^2 is dropped.
// ---------------------------------------------------------------------------
__global__ __launch_bounds__(128) void vq_main(const float* __restrict__ lat,
                                               const float* __restrict__ emb,
                                               const float* __restrict__ enorm_g,
                                               float* __restrict__ out,
                                               float* __restrict__ partial) {
    __shared__ float tile[TILE_PX * CH_STRIDE];  // latents [pixel][channel], padded
    __shared__ float ebuf[2][CT_FLOATS];         // double-buffered codebook tile
    __shared__ float enorm[K_CODES];
    __shared__ int   inds[TILE_PX];
    __shared__ float lsum[128];

    const int tid  = threadIdx.x;
    const int lane = tid & 31;
    const int wave = tid >> 5;
    // readfirstlane forces a wave-uniform SGPR -> scalar branch around the TDM
    // issue (TDM ignores EXEC, so EXEC-masking would not suppress it).
    const int wave0 = (__builtin_amdgcn_readfirstlane(tid) < 32);

    const int tileId = blockIdx.x;               // 0..4095
    const int b      = tileId >> 6;
    const int hw0    = (tileId & 63) * TILE_PX;  // one full h-row of 64 w's
    const float* latBase = lat + (size_t)b * D_DIM * HWPLANE + hw0;
    float*       outBase = out + (size_t)b * D_DIM * HWPLANE + hw0;

    // --- prologue: kick TDM for code tile 0 into ebuf[0] --------------------
    if (wave0) {
        tdm_load_tile(emb, (unsigned)(unsigned long long)&ebuf[0][0]);
    }

    // --- stage latents [64ch x 64px] -> LDS transposed to [px][ch] ----------
    {
        int p  = tid & 63;
        int c0 = tid >> 6;                        // 0 or 1
        for (int c = c0; c < D_DIM; c += 2)
            tile[p * CH_STRIDE + c] = latBase[(size_t)c * HWPLANE + p];
    }
    for (int k = tid; k < K_CODES; k += 128) enorm[k] = enorm_g[k];

    if (wave0) __builtin_amdgcn_s_wait_tensorcnt(0);
    __syncthreads();                              // tile, enorm, ebuf[0] ready

    // --- per-wave WMMA: 16 pixels vs all 512 codes --------------------------
    const int ml = lane & 15;                     // row/col index within tile
    const int kk = (lane >> 4) * 2;               // K sub-offset per lane half
    const int px = wave * 16 + ml;

    // Hoist A fragments (16 k-steps cover D=64; A is 16x4 f32 per WMMA).
    // ISA layout: VGPR0 = A[m][kk], VGPR1 = A[m][kk+1], m = lane%16.
    v2f afrag[16];
#pragma unroll
    for (int ks = 0; ks < 16; ++ks)
        afrag[ks] = *(const v2f*)&tile[px * CH_STRIDE + 4 * ks + kk];

    float bestv[8];
    int   besti[8];
#pragma unroll
    for (int r = 0; r < 8; ++r) { bestv[r] = 3.4e38f; besti[r] = 0; }

    for (int ct = 0; ct < 32; ++ct) {             // 32 code tiles of 16 codes
        const int buf = ct & 1;
        // overlap: DMA next tile into the other buffer while we compute
        if (wave0 && (ct + 1 < 32)) {
            tdm_load_tile(emb + (size_t)(ct + 1) * CT_FLOATS,
                          (unsigned)(unsigned long long)&ebuf[(ct + 1) & 1][0]);
        }

        const int ncode = ct * 16 + ml;           // this lane's B column
        const float* eb = &ebuf[buf][ml * D_DIM + kk];
        v8f acc = {};
#pragma unroll
        for (int ks = 0; ks < 16; ++ks) {
            // B 4x16: VGPR0 = E[n][4ks+kk], VGPR1 = E[n][4ks+kk+1]  (LDS)
            v2f bfrag = *(const v2f*)(eb + 4 * ks);
            acc = __builtin_amdgcn_wmma_f32_16x16x4_f32(
                false, afrag[ks], false, bfrag, (short)0, acc, false, false);
        }
        // acc[r] = S[m = r + 8*(lane/16)][n = lane%16]
        const float en = enorm[ncode];
#pragma unroll
        for (int r = 0; r < 8; ++r) {
            float dv = en - 2.0f * acc[r];
            if (dv < bestv[r]) { bestv[r] = dv; besti[r] = ncode; }
        }

        if (wave0 && (ct + 1 < 32)) __builtin_amdgcn_s_wait_tensorcnt(0);
        __syncthreads();                          // next buffer ready, old free
    }

    // --- cross-lane argmin within each 16-lane half (wave32 butterflies) ----
#pragma unroll
    for (int r = 0; r < 8; ++r) {
        float bv = bestv[r];
        int   bi = besti[r];
#pragma unroll
        for (int mask = 1; mask < 16; mask <<= 1) {
            float ov = __shfl_xor(bv, mask, 32);
            int   oi = __shfl_xor(bi, mask, 32);
            if (ov < bv || (ov == bv && oi < bi)) { bv = ov; bi = oi; }
        }
        if ((lane & 15) == 0) {
            int mr = r + ((lane >> 4) << 3);      // D-matrix row -> pixel
            inds[wave * 16 + mr] = bi;
        }
    }
    __syncthreads();

    // --- gather codebook, straight-through output, loss partial -------------
    {
        int   p     = tid & 63;
        int   dBase = (tid >> 6) * 32;            // 0 or 32
        int   ind   = inds[p];
        const float* e = emb + (size_t)ind * D_DIM;
        float s = 0.f;
#pragma unroll
        for (int j = 0; j < 32; ++j) {
            int   d    = dBase + j;
            float x    = tile[p * CH_STRIDE + d];
            float diff = e[d] - x;                // quant - lat (fp32, as ref)
            outBase[(size_t)d * HWPLANE + p] = x + diff;   // straight-through
            s += diff * diff;
        }
        lsum[tid] = s;
    }
    __syncthreads();
    if (tid == 0) {
        float t = 0.f;
        for (int i = 0; i < 128; ++i) t += lsum[i];        // fixed order
        partial[blockIdx.x] = t;
    }
}

// ---------------------------------------------------------------------------
// Kernel 3: deterministic fixed-order loss reduction.
// ---------------------------------------------------------------------------
__global__ __launch_bounds__(128) void vq_loss_reduce(const float* __restrict__ partial,
                                                      float* __restrict__ out) {
    __shared__ float lsum[128];
    float s = 0.f;
    for (int i = threadIdx.x; i < N_TILES; i += 128) s += partial[i];
    lsum[threadIdx.x] = s;
    __syncthreads();
    if (threadIdx.x == 0) {
        float t = 0.f;
        for (int i = 0; i < 128; ++i) t += lsum[i];
        // vq_loss = (BETA + 1) * mean((quant - lat)^2), N*D = 16777216
        out[16777216] = t * (1.25f / 16777216.0f);
    }
}

// ---------------------------------------------------------------------------
extern "C" void kernel_launch(void* const* d_in, const int* in_sizes, int n_in,
                              void* d_out, int out_size, void* d_ws, size_t ws_size,
                              hipStream_t stream) {
    const float* lat = (const float*)d_in[0];    // [64,64,64,64] fp32
    const float* emb = (const float*)d_in[1];    // [512,64] fp32
    float* out = (float*)d_out;                  // 64*64*64*64 quant + 1 loss
    float* ws  = (float*)d_ws;
    float* enorm   = ws;                         // 512 floats
    float* partial = ws + K_CODES;               // 4096 floats

    vq_prep<<<2, 256, 0, stream>>>(emb, enorm);
    vq_main<<<N_TILES, 128, 0, stream>>>(lat, emb, enorm, out, partial);
    vq_loss_reduce<<<1, 128, 0, stream>>>(partial, out);
}